// MambaTimeSeriesModel_88819923681541
// MI455X (gfx1250) — compile-verified
//
#include <hip/hip_runtime.h>
#include <hip/hip_bf16.h>

#define L_      4
#define D_      256
#define H_      512
#define N_      16
#define R_      8
#define T_      1024
#define B_      8
#define IN_DIM_ 32
#define OUT_LEN_ 96
#define MR      (B_ * T_)   /* 8192 rows of (b,t) */

#define ACT_NONE     0
#define ACT_RELU     1
#define ACT_SOFTPLUS 2

typedef float v2f __attribute__((ext_vector_type(2)));
typedef float v8f __attribute__((ext_vector_type(8)));

// ---------------------------------------------------------------------------
// Generic WMMA f32 GEMM: out[m][n] = act(sum_k A[m][k]*W[n][k] + bias[n])
//                                    (+ positional-encoding) (+ res[m][n])
// A: M x K row-major, W: N x K row-major (weights stored (out,in)).
// One wave per 16x16 output tile; V_WMMA_F32_16X16X4_F32 stepping K by 4.
// Columns n >= N read a clamped W row; their results are never stored, and
// since D[:,n] depends only on B[:,n] they cannot pollute valid columns.
// ---------------------------------------------------------------------------
__global__ void wmma_gemm_kernel(const float* __restrict__ A,
                                 const float* __restrict__ W,
                                 const float* __restrict__ bias,
                                 const float* __restrict__ res,
                                 float* __restrict__ out,
                                 int M, int N, int K,
                                 int act, int add_pe)
{
    const int lane   = threadIdx.x & 31;
    const int wib    = threadIdx.x >> 5;
    const int tileId = blockIdx.x * (blockDim.x >> 5) + wib;
    const int ntn    = (N + 15) >> 4;
    const int tm     = tileId / ntn;
    const int tn     = tileId % ntn;
    if (tm * 16 >= M) return;                 // uniform per wave

    const int hv = lane >> 4;                 // half-wave select (K pair)
    const int mn = lane & 15;
    const int m  = tm * 16 + mn;              // A row this lane feeds
    const int n  = tn * 16 + mn;              // out column / W row
    const int nc = (n < N) ? n : (N - 1);     // clamp only; no value masking

    const float* __restrict__ Arow = A + (size_t)m  * K;
    const float* __restrict__ Wrow = W + (size_t)nc * K;

    v8f c = {};
#pragma unroll 8
    for (int k0 = 0; k0 < K; k0 += 4) {
        const int kb = k0 + hv * 2;
        v2f a, b;
        a.x = Arow[kb];
        a.y = Arow[kb + 1];
        b.x = Wrow[kb];
        b.y = Wrow[kb + 1];
        c = __builtin_amdgcn_wmma_f32_16x16x4_f32(false, a, false, b,
                                                  (short)0, c, false, false);
    }

    if (n >= N) return;                       // dead columns: no store
    const float bv = (bias != nullptr) ? bias[n] : 0.0f;

#pragma unroll
    for (int r = 0; r < 8; ++r) {
        const int mrow = tm * 16 + r + hv * 8;
        float v = c[r] + bv;
        if (act == ACT_RELU) {
            v = fmaxf(v, 0.0f);
        } else if (act == ACT_SOFTPLUS) {
            v = (v > 20.0f) ? v : log1pf(__expf(v));
        }
        if (add_pe) {
            const int  t    = mrow & (T_ - 1);
            const float frq = __expf(-(float)(n & ~1) *
                                     (9.210340371976184f / (float)D_));
            const float ang = (float)t * frq;
            v += (n & 1) ? __cosf(ang) : __sinf(ang);
        }
        const size_t o = (size_t)mrow * N + n;
        if (res != nullptr) v += res[o];
        out[o] = v;
    }
}

// ---------------------------------------------------------------------------
// LayerNorm over D=256, one wave per row (8 elems/lane, wave32 reduction)
// ---------------------------------------------------------------------------
__global__ void layernorm_kernel(const float* __restrict__ x,
                                 const float* __restrict__ g,
                                 const float* __restrict__ b,
                                 float* __restrict__ out, int rows)
{
    const int lane = threadIdx.x & 31;
    const int row  = (blockIdx.x * blockDim.x + threadIdx.x) >> 5;
    if (row >= rows) return;
    const float* xr = x + (size_t)row * D_;

    float vals[8];
    float s = 0.0f;
#pragma unroll
    for (int i = 0; i < 8; ++i) { vals[i] = xr[lane + 32 * i]; s += vals[i]; }
#pragma unroll
    for (int msk = 16; msk >= 1; msk >>= 1) s += __shfl_xor(s, msk, 32);
    const float mean = s * (1.0f / D_);

    float vs = 0.0f;
#pragma unroll
    for (int i = 0; i < 8; ++i) { const float d = vals[i] - mean; vs += d * d; }
#pragma unroll
    for (int msk = 16; msk >= 1; msk >>= 1) vs += __shfl_xor(vs, msk, 32);
    const float rstd = rsqrtf(vs * (1.0f / D_) + 1e-5f);

    float* orow = out + (size_t)row * D_;
#pragma unroll
    for (int i = 0; i < 8; ++i) {
        const int d = lane + 32 * i;
        orow[d] = (vals[i] - mean) * rstd * g[d] + b[d];
    }
}

// ---------------------------------------------------------------------------
// Depthwise conv (k=3, pad=1) along t + SiLU. v layout (b,t,h).
// ---------------------------------------------------------------------------
__global__ void conv_silu_kernel(const float* __restrict__ v,
                                 const float* __restrict__ cw,  // (H,3)
                                 float* __restrict__ out)
{
    const int idx = blockIdx.x * blockDim.x + threadIdx.x;
    if (idx >= B_ * T_ * H_) return;
    const int h  = idx % H_;
    const int t  = (idx / H_) % T_;
    const float w0 = cw[h * 3 + 0], w1 = cw[h * 3 + 1], w2 = cw[h * 3 + 2];
    const float xm = (t > 0)      ? v[idx - H_] : 0.0f;
    const float xc = v[idx];
    const float xp = (t < T_ - 1) ? v[idx + H_] : 0.0f;
    const float a  = w0 * xm + w1 * xc + w2 * xp;
    out[idx] = a / (1.0f + __expf(-a));       // SiLU
}

// ---------------------------------------------------------------------------
// Selective scan: one 16-lane half-wave per (b,h) channel, lane = n state.
// Writes y in-place over xv (safe: slot owned by this half-wave only).
// ---------------------------------------------------------------------------
__global__ void scan_kernel(const float* __restrict__ xv,   // (b,t,h)
                            const float* __restrict__ dt,   // (b,t,h)
                            const float* __restrict__ A_p,  // (H,N)
                            const float* __restrict__ D_p,  // (H)
                            float* __restrict__ y)          // (b,t,h)
{
    const int lane = threadIdx.x & 31;
    const int n    = lane & 15;
    const int ch   = (blockIdx.x * blockDim.x + threadIdx.x) >> 4;
    if (ch >= B_ * H_) return;
    const int b = ch / H_;
    const int h = ch % H_;

    const float Ap    = A_p[h * N_ + n];
    const float Aneg  = -__expf(Ap);
    const float safeA = (fabsf(Aneg) < 1e-5f) ? 1.0f : Aneg;
    const float Dp    = D_p[h];

    float state = 0.0f;
    size_t base = (size_t)b * T_ * H_ + h;
    for (int t = 0; t < T_; ++t) {
        const float x_t   = xv[base];
        const float d_t   = dt[base];
        const float discA = __expf(Aneg * d_t);
        const float discB = (discA - 1.0f) / safeA * d_t;
        state = discA * state + discB * x_t;
        float contrib = state * Ap;
        contrib += __shfl_xor(contrib, 1, 32);
        contrib += __shfl_xor(contrib, 2, 32);
        contrib += __shfl_xor(contrib, 4, 32);
        contrib += __shfl_xor(contrib, 8, 32);
        if (n == 0) y[base] = contrib + Dp * x_t;
        base += H_;
    }
}

// ---------------------------------------------------------------------------
// Final LayerNorm of h[:, -1, :] -> (B, D). One wave per batch row.
// ---------------------------------------------------------------------------
__global__ void final_ln_kernel(const float* __restrict__ h,
                                const float* __restrict__ g,
                                const float* __restrict__ b,
                                float* __restrict__ out)
{
    const int lane = threadIdx.x & 31;
    const int wv   = threadIdx.x >> 5;
    if (wv >= B_) return;
    const float* xr = h + ((size_t)wv * T_ + (T_ - 1)) * D_;

    float vals[8];
    float s = 0.0f;
#pragma unroll
    for (int i = 0; i < 8; ++i) { vals[i] = xr[lane + 32 * i]; s += vals[i]; }
#pragma unroll
    for (int msk = 16; msk >= 1; msk >>= 1) s += __shfl_xor(s, msk, 32);
    const float mean = s * (1.0f / D_);

    float vs = 0.0f;
#pragma unroll
    for (int i = 0; i < 8; ++i) { const float d = vals[i] - mean; vs += d * d; }
#pragma unroll
    for (int msk = 16; msk >= 1; msk >>= 1) vs += __shfl_xor(vs, msk, 32);
    const float rstd = rsqrtf(vs * (1.0f / D_) + 1e-5f);

#pragma unroll
    for (int i = 0; i < 8; ++i) {
        const int d = lane + 32 * i;
        out[wv * D_ + d] = (vals[i] - mean) * rstd * g[d] + b[d];
    }
}

// ---------------------------------------------------------------------------
// Tiny head GEMM (M = B_ rows), scalar: out[m][n] = x[m]·W[n] + bias (opt SiLU)
// ---------------------------------------------------------------------------
__global__ void head_kernel(const float* __restrict__ x,   // (B, K)
                            const float* __restrict__ W,   // (N, K)
                            const float* __restrict__ bias,
                            float* __restrict__ out, int N, int K, int silu)
{
    const int idx = blockIdx.x * blockDim.x + threadIdx.x;
    if (idx >= B_ * N) return;
    const int m = idx / N, n = idx % N;
    const float* xr = x + (size_t)m * K;
    const float* wr = W + (size_t)n * K;
    float s = (bias != nullptr) ? bias[n] : 0.0f;
    for (int k = 0; k < K; ++k) s += xr[k] * wr[k];
    if (silu) s = s / (1.0f + __expf(-s));
    out[idx] = s;
}

// ---------------------------------------------------------------------------
extern "C" void kernel_launch(void* const* d_in, const int* in_sizes, int n_in,
                              void* d_out, int out_size, void* d_ws, size_t ws_size,
                              hipStream_t stream)
{
    const float* x       = (const float*)d_in[0];
    const float* emb_w   = (const float*)d_in[1];
    const float* emb_b   = (const float*)d_in[2];
    const float* norm_g  = (const float*)d_in[3];
    const float* norm_b  = (const float*)d_in[4];
    const float* in_w    = (const float*)d_in[5];
    const float* conv_w  = (const float*)d_in[6];
    const float* A_p     = (const float*)d_in[7];
    const float* D_p     = (const float*)d_in[8];
    const float* dt1_w   = (const float*)d_in[9];
    const float* dt1_b   = (const float*)d_in[10];
    const float* dt2_w   = (const float*)d_in[11];
    const float* dt2_b   = (const float*)d_in[12];
    const float* out_w   = (const float*)d_in[13];
    const float* final_g = (const float*)d_in[14];
    const float* final_b = (const float*)d_in[15];
    const float* head1_w = (const float*)d_in[16];
    const float* head1_b = (const float*)d_in[17];
    const float* head2_w = (const float*)d_in[18];
    const float* head2_b = (const float*)d_in[19];
    float* out = (float*)d_out;

    // Workspace layout (floats). Total ~48.3 MiB.
    float* ws  = (float*)d_ws;
    float* h   = ws;                         // MR*D   residual stream
    float* ln  = h   + (size_t)MR * D_;      // MR*D   normalized h
    float* v   = ln  + (size_t)MR * D_;      // MR*H   in_proj out; reused: dt full
    float* vc  = v   + (size_t)MR * H_;      // MR*H   conv+silu out; reused: y (in place)
    float* dtr = vc  + (size_t)MR * H_;      // MR*R   dt rank-space
    float* hl  = dtr + (size_t)MR * R_;      // B*D    final LN of last step
    float* t1  = hl  + (size_t)B_ * D_;      // B*D    head1 out

    // Embedding GEMM + bias + positional encoding: h = x·emb_w^T + emb_b + pe
    {
        const int tiles = (MR / 16) * (D_ / 16);
        wmma_gemm_kernel<<<(tiles + 7) / 8, 256, 0, stream>>>(
            x, emb_w, emb_b, nullptr, h, MR, D_, IN_DIM_, ACT_NONE, 1);
    }

    for (int l = 0; l < L_; ++l) {
        // LayerNorm
        layernorm_kernel<<<MR / 8, 256, 0, stream>>>(
            h, norm_g + l * D_, norm_b + l * D_, ln, MR);
        // in_proj: v = ln · in_w[l]^T   (8192x256 @ 256x512)
        {
            const int tiles = (MR / 16) * (H_ / 16);
            wmma_gemm_kernel<<<(tiles + 7) / 8, 256, 0, stream>>>(
                ln, in_w + (size_t)l * H_ * D_, nullptr, nullptr, v,
                MR, H_, D_, ACT_NONE, 0);
        }
        // depthwise conv3 + SiLU
        conv_silu_kernel<<<(MR * H_) / 256, 256, 0, stream>>>(
            v, conv_w + (size_t)l * H_ * 3, vc);
        // dt rank projection + relu  (N=8, guarded tile)
        {
            const int tiles = (MR / 16) * 1;
            wmma_gemm_kernel<<<(tiles + 7) / 8, 256, 0, stream>>>(
                vc, dt1_w + (size_t)l * R_ * H_, dt1_b + l * R_, nullptr, dtr,
                MR, R_, H_, ACT_RELU, 0);
        }
        // dt expand + softplus (K=8), result overwrites v
        {
            const int tiles = (MR / 16) * (H_ / 16);
            wmma_gemm_kernel<<<(tiles + 7) / 8, 256, 0, stream>>>(
                dtr, dt2_w + (size_t)l * H_ * R_, dt2_b + l * H_, nullptr, v,
                MR, H_, R_, ACT_SOFTPLUS, 0);
        }
        // selective scan: y written in place over vc
        scan_kernel<<<(B_ * H_ * 16) / 256, 256, 0, stream>>>(
            vc, v, A_p + (size_t)l * H_ * N_, D_p + (size_t)l * H_, vc);
        // out_proj + residual (in place on h)
        {
            const int tiles = (MR / 16) * (D_ / 16);
            wmma_gemm_kernel<<<(tiles + 7) / 8, 256, 0, stream>>>(
                vc, out_w + (size_t)l * D_ * H_, nullptr, h, h,
                MR, D_, H_, ACT_NONE, 0);
        }
    }

    // Final LN on last timestep, then heads
    final_ln_kernel<<<1, 256, 0, stream>>>(h, final_g, final_b, hl);
    head_kernel<<<(B_ * D_ + 255) / 256, 256, 0, stream>>>(
        hl, head1_w, head1_b, t1, D_, D_, 1);
    head_kernel<<<(B_ * OUT_LEN_ + 255) / 256, 256, 0, stream>>>(
        t1, head2_w, head2_b, out, OUT_LEN_, D_, 0);
}